// Mu_data_66786741453413
// MI455X (gfx1250) — compile-verified
//
#include <hip/hip_runtime.h>

typedef __attribute__((ext_vector_type(2))) float v2f;
typedef __attribute__((ext_vector_type(8))) float v8f;

namespace {
constexpr int kB = 64;          // batches
constexpr int kK = 4;           // prediction channels
constexpr int kM = 3;           // input channels
constexpr int kP = 256 * 256;   // pixels per plane
constexpr int kSplit = 8;       // blocks per batch
constexpr int kThreads = 256;   // 8 wave32 per block
constexpr float kEps = 1e-10f;
}

// --- kernel 0: zero the per-batch accumulators (B x 16 floats in d_ws) ----
__global__ void zero_kernel(float* ws, int n) {
  int i = blockIdx.x * blockDim.x + threadIdx.x;
  if (i < n) ws[i] = 0.0f;
}

// --- kernel 1: bandwidth-bound streaming pass -----------------------------
// ws[b*16 + k*4 + m] = num[b,k,m] for m<3 ; ws[b*16 + k*4 + 3] = den[b,k]
__global__ __launch_bounds__(kThreads) void batch_sums_kernel(
    const float* __restrict__ pred, const float* __restrict__ inp,
    const int* __restrict__ heart, float* __restrict__ ws) {
  const int b = blockIdx.y;
  const int chunk = blockIdx.x;
  const int t = threadIdx.x;
  const int pxPerBlock = kP / kSplit;       // 8192 pixels
  const int vecPerBlock = pxPerBlock / 4;   // 2048 float4 elements
  const int basePx = chunk * pxPerBlock;

  const float4* pp[kK];
#pragma unroll
  for (int k = 0; k < kK; ++k)
    pp[k] = reinterpret_cast<const float4*>(pred + ((size_t)b * kK + k) * kP + basePx);
  const float4* ip[kM];
#pragma unroll
  for (int m = 0; m < kM; ++m)
    ip[m] = reinterpret_cast<const float4*>(inp + ((size_t)b * kM + m) * kP + basePx);
  const int4* hp = reinterpret_cast<const int4*>(heart + (size_t)b * kP + basePx);

  float acc[16];
#pragma unroll
  for (int i = 0; i < 16; ++i) acc[i] = 0.0f;

  for (int i = t; i < vecPerBlock; i += kThreads) {
    const int4 h = hp[i];
    const float4 mk = make_float4(h.x == 1 ? 1.f : 0.f, h.y == 1 ? 1.f : 0.f,
                                  h.z == 1 ? 1.f : 0.f, h.w == 1 ? 1.f : 0.f);
    float4 im[kM];
#pragma unroll
    for (int m = 0; m < kM; ++m) im[m] = ip[m][i];
#pragma unroll
    for (int k = 0; k < kK; ++k) {
      const float4 p = pp[k][i];
      const float4 pm = make_float4(p.x * mk.x, p.y * mk.y, p.z * mk.z, p.w * mk.w);
      acc[k * 4 + 3] += (pm.x + pm.y) + (pm.z + pm.w);
#pragma unroll
      for (int m = 0; m < kM; ++m) {
        acc[k * 4 + m] = fmaf(pm.x, im[m].x,
                         fmaf(pm.y, im[m].y,
                         fmaf(pm.z, im[m].z,
                         fmaf(pm.w, im[m].w, acc[k * 4 + m]))));
      }
    }
  }

  // wave32 butterfly reduction (warpSize == 32 on gfx1250)
#pragma unroll
  for (int i = 0; i < 16; ++i) {
#pragma unroll
    for (int off = 16; off > 0; off >>= 1)
      acc[i] += __shfl_xor(acc[i], off, 32);
  }

  __shared__ float red[kThreads / 32][16];
  const int wave = t >> 5;
  if ((t & 31) == 0) {
#pragma unroll
    for (int i = 0; i < 16; ++i) red[wave][i] = acc[i];
  }
  __syncthreads();
  if (t < 16) {
    float s = 0.f;
#pragma unroll
    for (int w = 0; w < kThreads / 32; ++w) s += red[w][t];
    atomicAdd(&ws[b * 16 + t], s);
  }
}

// --- kernel 2: one wave32; mu_mean via chained V_WMMA_F32_16X16X4_F32 -----
// A (16x4): row r = k*4+m slot, contraction dim = 4 batches per WMMA.
//   Documented layout: lanes 0-15 row=lane, VGPR0->K=0, VGPR1->K=1;
//   lanes 16-31 row=lane-16, VGPR0->K=2, VGPR1->K=3.
// B = all-ones (layout-independent).  D accumulates Sum_b num/den/B.
__global__ void finalize_kernel(const float* __restrict__ ws,
                                const float* __restrict__ mu_data,
                                float* __restrict__ out) {
  const int lane = threadIdx.x;     // 0..31, EXEC all ones
  const int r = lane & 15;          // row = k*4 + m   (m==3 rows are zero)
  const int half = lane >> 4;
  const int k = r >> 2;
  const int m = r & 3;
  const float invB = 1.0f / (float)kB;

  v8f c = {0.f, 0.f, 0.f, 0.f, 0.f, 0.f, 0.f, 0.f};
  const v2f onesv = {1.0f, 1.0f};

#pragma unroll
  for (int j = 0; j < kB / 4; ++j) {        // 16 chained WMMAs over 64 batches
    const int b0 = 4 * j + (half ? 2 : 0);  // K-slot 0/2 for this lane half
    const int b1 = b0 + 1;                  // K-slot 1/3
    const float n0 = ws[b0 * 16 + r];
    const float d0 = ws[b0 * 16 + k * 4 + 3] + kEps;
    const float n1 = ws[b1 * 16 + r];
    const float d1 = ws[b1 * 16 + k * 4 + 3] + kEps;
    v2f a;
    a.x = (m == 3) ? 0.0f : (n0 / d0) * invB;
    a.y = (m == 3) ? 0.0f : (n1 / d1) * invB;
    c = __builtin_amdgcn_wmma_f32_16x16x4_f32(
        /*neg_a=*/false, a, /*neg_b=*/false, onesv,
        /*c_mod=*/(short)0, c, /*reuse_a=*/false, /*reuse_b=*/false);
  }

  // D layout: lanes<16 hold rows 0..7 in c[0..7]; lanes>=16 hold rows 8..15.
  // All columns identical (B = ones), so every lane in a half has the data.
  const int kbase = half * 2;
  float part = 0.0f;
#pragma unroll
  for (int v = 0; v < 8; ++v) {
    const int mm = v & 3;
    if (mm == 3) continue;                  // den rows were zeroed in A
    const int kk = kbase + (v >> 2);
    const float diff = mu_data[kk * kM + mm] - c[v];
    part = fmaf(diff, diff, part);
  }
  const float total = part + __shfl_xor(part, 16, 32);
  if (lane == 0) out[0] = total;
}

extern "C" void kernel_launch(void* const* d_in, const int* in_sizes, int n_in,
                              void* d_out, int out_size, void* d_ws, size_t ws_size,
                              hipStream_t stream) {
  const float* pred    = (const float*)d_in[0];  // [64,4,256,256] f32
  const float* inp     = (const float*)d_in[1];  // [64,3,256,256] f32
  const int*   heart   = (const int*)d_in[2];    // [64,1,256,256] i32
  const float* mu_data = (const float*)d_in[3];  // [4,3] f32
  float* out = (float*)d_out;                    // scalar
  float* ws  = (float*)d_ws;                     // 64*16 floats

  const int nAcc = kB * 16;
  zero_kernel<<<(nAcc + 255) / 256, 256, 0, stream>>>(ws, nAcc);
  batch_sums_kernel<<<dim3(kSplit, kB), kThreads, 0, stream>>>(pred, inp, heart, ws);
  finalize_kernel<<<1, 32, 0, stream>>>(ws, mu_data, out);
}